// HighOrderAggregator_26740466385630
// MI455X (gfx1250) — compile-verified
//
#include <hip/hip_runtime.h>

typedef __attribute__((ext_vector_type(2))) float v2f;
typedef __attribute__((ext_vector_type(8))) float v8f;

#define DFEAT 64
#define EPS_NORM 1e-9f

// ---------------------------------------------------------------------------
// Kernel 1: zero the aggregation workspace (harness poisons d_ws; we must
// re-zero every call since SpMM accumulates with atomics).
// ---------------------------------------------------------------------------
__global__ void zero_f32_kernel(float* __restrict__ p, int n) {
  int i = blockIdx.x * blockDim.x + threadIdx.x;
  if (i < n) p[i] = 0.0f;
}

// ---------------------------------------------------------------------------
// Kernel 2: SpMM scatter  agg[row[e]] += val[e] * x[col[e]]
// One thread per (edge, feature). x (25.6 MB) and agg (25.6 MB) are
// L2-resident on MI455X (192 MB L2), so gathers and f32 atomics stay in L2.
// ---------------------------------------------------------------------------
__global__ __launch_bounds__(256)
void spmm_scatter_kernel(const float* __restrict__ x,
                         const float* __restrict__ edge_val,
                         const int* __restrict__ edge_row,
                         const int* __restrict__ edge_col,
                         float* __restrict__ agg,
                         int n_edges) {
  long long t = (long long)blockIdx.x * 256 + threadIdx.x;
  int e = (int)(t >> 6);
  if (e >= n_edges) return;
  int f = (int)(t & 63);
  int r = edge_row[e];
  int c = edge_col[e];
  float v = edge_val[e] * x[(long long)c * DFEAT + f];
  atomicAdd(agg + (long long)r * DFEAT + f, v);
}

// ---------------------------------------------------------------------------
// Fused hop transform for one wave's 16-row block:
//   h = relu(feat @ W^T + b);  res = (h - mean)*scale*rsqrt(var+eps) + offset
// GEMM via V_WMMA_F32_16X16X4_F32 (f32 in/out, exact precision).
//
// A-matrix (16x4 f32) layout per ISA: lane = {M=lane&15}, VGPR pair holds
// K = 4k + 2*(lane>=16) and +1.  B-matrix (4x16) mirrors it with
// B[k][n] = W[n][k], n = lane&15.  C/D: VGPR j -> row j + 8*(lane>=16),
// col = lane&15, so per-row reductions are xor-shuffles over the 16-lane half.
// ---------------------------------------------------------------------------
__device__ __forceinline__ void hop_transform_tiles(
    const float* __restrict__ frow,   // this lane's (clamped) feature row
    const float* ldsW,                // [64x64] W row-major (D_OUT x D_IN) in LDS
    const float* __restrict__ bias,
    const float* __restrict__ scale,
    const float* __restrict__ offset,
    int half, int l16, v8f res[4])
{
  // Preload this lane's 32 A-values (reused across all 4 N-tiles).
  v2f A[16];
#pragma unroll
  for (int k = 0; k < 16; ++k) {
    A[k] = *(const v2f*)(frow + k * 4 + half * 2);
  }
  float bv[4], sv[4], ov[4];
#pragma unroll
  for (int t = 0; t < 4; ++t) {
    int n = t * 16 + l16;
    bv[t] = bias[n];
    sv[t] = scale[n];
    ov[t] = offset[n];
  }
  // 4 N-tiles x 16 K-steps of f32 WMMA.
#pragma unroll
  for (int t = 0; t < 4; ++t) {
    const float* wrow = ldsW + (t * 16 + l16) * DFEAT + half * 2;
    v8f c = {};
#pragma unroll
    for (int k = 0; k < 16; ++k) {
      v2f b = *(const v2f*)(wrow + k * 4);   // ds_load_b64
      c = __builtin_amdgcn_wmma_f32_16x16x4_f32(
              /*neg_a=*/false, A[k], /*neg_b=*/false, b,
              /*c_mod=*/(short)0, c, /*reuse_a=*/false, /*reuse_b=*/false);
    }
    res[t] = c;
  }
  // Bias + ReLU + per-row normalization, fully in registers.
#pragma unroll
  for (int j = 0; j < 8; ++j) {
    float s1 = 0.0f, s2 = 0.0f;
#pragma unroll
    for (int t = 0; t < 4; ++t) {
      float h = res[t][j] + bv[t];
      h = fmaxf(h, 0.0f);
      res[t][j] = h;
      s1 += h;
      s2 += h * h;
    }
    // Reduce across the 16 lanes of this half-wave (rows stay in-half).
#pragma unroll
    for (int m = 1; m < 16; m <<= 1) {
      s1 += __shfl_xor(s1, m, 32);
      s2 += __shfl_xor(s2, m, 32);
    }
    float mean = s1 * (1.0f / DFEAT);
    float var  = s2 * (1.0f / DFEAT) - mean * mean + EPS_NORM;
    float rinv = rsqrtf(var);
#pragma unroll
    for (int t = 0; t < 4; ++t) {
      res[t][j] = (res[t][j] - mean) * sv[t] * rinv + ov[t];
    }
  }
}

// ---------------------------------------------------------------------------
// Kernel 3: fused dual-hop transform + sum.
// 256 threads = 8 waves; each wave owns a 16-row block -> 128 rows / block.
// W0,W1 (32 KB) staged in LDS once per block.
// ---------------------------------------------------------------------------
__global__ __launch_bounds__(256)
void gcn_fused_kernel(const float* __restrict__ x,
                      const float* __restrict__ agg,
                      const float* __restrict__ W0, const float* __restrict__ b0,
                      const float* __restrict__ s0, const float* __restrict__ o0,
                      const float* __restrict__ W1, const float* __restrict__ b1,
                      const float* __restrict__ s1, const float* __restrict__ o1,
                      float* __restrict__ out, int n_nodes)
{
  __shared__ float ldsW0[DFEAT * DFEAT];
  __shared__ float ldsW1[DFEAT * DFEAT];
  for (int i = threadIdx.x; i < DFEAT * DFEAT; i += 256) {
    ldsW0[i] = W0[i];
    ldsW1[i] = W1[i];
  }
  __syncthreads();

  const int lane = threadIdx.x & 31;
  const int wave = threadIdx.x >> 5;
  const int half = lane >> 4;
  const int l16  = lane & 15;
  const int row_base = (blockIdx.x * 8 + wave) * 16;

  // Clamp load row so EXEC stays all-1s through the WMMA region.
  int r = row_base + l16;
  r = (r < n_nodes) ? r : (n_nodes - 1);

  v8f h0[4], h1[4];
  hop_transform_tiles(x   + (long long)r * DFEAT, ldsW0, b0, s0, o0, half, l16, h0);
  hop_transform_tiles(agg + (long long)r * DFEAT, ldsW1, b1, s1, o1, half, l16, h1);

#pragma unroll
  for (int t = 0; t < 4; ++t) {
#pragma unroll
    for (int j = 0; j < 8; ++j) {
      int rr = row_base + j + half * 8;
      if (rr < n_nodes)
        out[(long long)rr * DFEAT + t * 16 + l16] = h0[t][j] + h1[t][j];
    }
  }
}

// ---------------------------------------------------------------------------
// Input order (setup_inputs): x, edge_val, W0, b0, scale0, offset0,
//                             W1, b1, scale1, offset1, edge_row, edge_col
// ---------------------------------------------------------------------------
extern "C" void kernel_launch(void* const* d_in, const int* in_sizes, int n_in,
                              void* d_out, int out_size, void* d_ws, size_t ws_size,
                              hipStream_t stream) {
  const float* x  = (const float*)d_in[0];
  const float* ev = (const float*)d_in[1];
  const float* W0 = (const float*)d_in[2];
  const float* b0 = (const float*)d_in[3];
  const float* s0 = (const float*)d_in[4];
  const float* o0 = (const float*)d_in[5];
  const float* W1 = (const float*)d_in[6];
  const float* b1 = (const float*)d_in[7];
  const float* s1 = (const float*)d_in[8];
  const float* o1 = (const float*)d_in[9];
  const int* er   = (const int*)d_in[10];
  const int* ec   = (const int*)d_in[11];
  float* out      = (float*)d_out;

  const int n_nodes = in_sizes[0] / DFEAT;
  const int n_edges = in_sizes[1];
  float* agg = (float*)d_ws;   // n_nodes * 64 floats (25.6 MB)

  const int agg_elems = n_nodes * DFEAT;
  zero_f32_kernel<<<(agg_elems + 255) / 256, 256, 0, stream>>>(agg, agg_elems);

  const long long sp_threads = (long long)n_edges * DFEAT;
  const int sp_blocks = (int)((sp_threads + 255) / 256);
  spmm_scatter_kernel<<<sp_blocks, 256, 0, stream>>>(x, ev, er, ec, agg, n_edges);

  const int tr_blocks = (n_nodes + 127) / 128;
  gcn_fused_kernel<<<tr_blocks, 256, 0, stream>>>(
      x, agg, W0, b0, s0, o0, W1, b1, s1, o1, out, n_nodes);
}